// DOF6Loss_61589831024942
// MI455X (gfx1250) — compile-verified
//
#include <hip/hip_runtime.h>
#include <hip/hip_bf16.h>

// ---------------------------------------------------------------------------
// DOF6 loss on MI455X (gfx1250, wave32).
// Dominant cost: row-wise sum-of-squares of prediction[:,1,:]  (64 MB stream)
//   -> pure HBM-bandwidth problem, ideal time ~64MB / 23.3TB/s ~= 2.9us.
// Per row: one 256-thread block, exactly 4 b128 loads per thread (fully
// unrolled, all 4 in flight), WMMA-based wave reduction
// (v_wmma_f32_16x16x4_f32 with B = ones), 6-element loss tail, then a
// deterministic single-block final reduction. No atomics -> bit-stable.
// ---------------------------------------------------------------------------

typedef __attribute__((ext_vector_type(2))) float v2f;
typedef __attribute__((ext_vector_type(8))) float v8f;

#define EPS_F 1e-9f
#define NORM_EPS_F 1e-12f

static constexpr int KDIM = 4096;   // reference shape (B, 2, 4096)
static constexpr int NT   = 256;    // threads per block (8 wave32 waves)
static constexpr int NW   = NT / 32;

// Full 32-lane sum using one f32 WMMA.
// A (16x4) holds all 32 lanes' {x, 0} in its 64 slots (exact lane->(m,k)
// mapping irrelevant since B is all-ones):  D[m,n] = sum_k A[m,k].
// Per-lane sum of D's 8 VGPRs = rows 0..7 (lanes 0-15) or rows 8..15
// (lanes 16-31); one shfl_xor(16) completes the reduction for every lane.
__device__ __forceinline__ float wave_sum_wmma(float x) {
    v2f a;    a[0] = x;    a[1] = 0.0f;
    v2f ones; ones[0] = 1.0f; ones[1] = 1.0f;
    v8f c = {};
    c = __builtin_amdgcn_wmma_f32_16x16x4_f32(
            /*neg_a=*/false, a, /*neg_b=*/false, ones,
            /*c_mod=*/(short)0, c, /*reuse_a=*/false, /*reuse_b=*/false);
    float s = c[0] + c[1] + c[2] + c[3] + c[4] + c[5] + c[6] + c[7];
    s += __shfl_xor(s, 16, 32);
    return s;
}

// Block-wide (256-thread) sum; valid in thread 0 only.
__device__ __forceinline__ float block_sum256(float x, float* lds) {
    const int lane = threadIdx.x & 31;
    const int wid  = threadIdx.x >> 5;
    float w = wave_sum_wmma(x);
    if (lane == 0) lds[wid] = w;
    __syncthreads();
    float total = 0.0f;
    if (threadIdx.x == 0) {
        #pragma unroll
        for (int i = 0; i < NW; ++i) total += lds[i];
    }
    return total;
}

// One block per row b: ssq of (p1 + EPS), then the 6-element loss tail.
__global__ void __launch_bounds__(NT)
dof6_row_kernel(const float* __restrict__ pred,
                const float* __restrict__ targ,
                float* __restrict__ partial) {
    const int b = blockIdx.x;
    const size_t rowBase = (size_t)b * (size_t)(2 * KDIM); // prediction[b,0,0]
    const float* p1 = pred + rowBase + KDIM;               // prediction[b,1,:]
    const float4* p1v = (const float4*)p1;                 // 16B-aligned
    const int t = threadIdx.x;

    // KDIM/4 = 1024 float4 per row, NT=256 threads -> exactly 4 each.
    // Issue all four b128 loads before consuming (max loads in flight).
    float4 v0 = p1v[t];
    float4 v1 = p1v[t + NT];
    float4 v2 = p1v[t + 2 * NT];
    float4 v3 = p1v[t + 3 * NT];

    float acc = 0.0f;
    #pragma unroll
    for (int j = 0; j < 4; ++j) {
        float4 v = (j == 0) ? v0 : (j == 1) ? v1 : (j == 2) ? v2 : v3;
        float a0 = v.x + EPS_F;
        float a1 = v.y + EPS_F;
        float a2 = v.z + EPS_F;
        float a3 = v.w + EPS_F;
        acc = fmaf(a0, a0, acc);
        acc = fmaf(a1, a1, acc);
        acc = fmaf(a2, a2, acc);
        acc = fmaf(a3, a3, acc);
    }

    __shared__ float lds[NW];
    float ssq = block_sum256(acc, lds);

    if (t == 0) {
        const float norm = sqrtf(ssq);
        const float inv  = 1.0f / fmaxf(norm, NORM_EPS_F);
        const float* p0 = pred + rowBase;   // prediction[b,0,:]
        const float* t0 = targ + rowBase;   // target[b,0,:]
        float s = 0.0f;
        #pragma unroll
        for (int k = 0; k < 6; ++k) {
            float p1n = (p1[k] + EPS_F) * inv;
            float p0e = p0[k] + EPS_F;
            p0e = (p1n < 0.5f) ? -p0e : p0e;
            float scale = (k < 3) ? 100.0f : 1000.0f;
            float d = p0e * scale - t0[k] * scale;
            s = fmaf(d, d, s);
        }
        partial[b] = s;   // row contribution to (sum_t + sum_r)
    }
}

// Single-block deterministic reduction of B partials -> scalar loss.
__global__ void __launch_bounds__(NT)
dof6_final_kernel(const float* __restrict__ partial,
                  float* __restrict__ out,
                  int B) {
    float s = 0.0f;
    for (int i = threadIdx.x; i < B; i += NT) s += partial[i];
    __shared__ float lds[NW];
    float total = block_sum256(s, lds);
    if (threadIdx.x == 0) {
        // translation_loss + rotation_loss, each a mean over B*3 elements.
        out[0] = total / (3.0f * (float)B);
    }
}

extern "C" void kernel_launch(void* const* d_in, const int* in_sizes, int n_in,
                              void* d_out, int out_size, void* d_ws, size_t ws_size,
                              hipStream_t stream) {
    const float* pred = (const float*)d_in[0];
    const float* targ = (const float*)d_in[1];
    float* out = (float*)d_out;
    float* ws  = (float*)d_ws;

    const int B = in_sizes[0] / (2 * KDIM);   // = 4096

    dof6_row_kernel<<<B, NT, 0, stream>>>(pred, targ, ws);
    dof6_final_kernel<<<1, NT, 0, stream>>>(ws, out, B);
}